// STGCNBlock_28157805592902
// MI455X (gfx1250) — compile-verified
//
#include <hip/hip_runtime.h>
#include <math.h>

// ---------------- problem constants ----------------
#define BB   16
#define CCH  64      // channels C
#define T0C  12
#define T1C  10      // after tconv1 (KT=3)
#define T2C  8       // after tconv2
#define NN   2048
#define NKC  6144    // KS * N
#define KCC  192     // C*KT == KS*C

typedef float v2f __attribute__((ext_vector_type(2)));
typedef float v8f __attribute__((ext_vector_type(8)));

typedef __attribute__((address_space(3))) void lds_void;

__device__ __forceinline__ v8f wmma_f32_k4(v2f a, v2f b, v8f c) {
  // V_WMMA_F32_16X16X4_F32 : D(16x16 f32) = A(16x4 f32) x B(4x16 f32) + C
  return __builtin_amdgcn_wmma_f32_16x16x4_f32(
      /*neg_a=*/false, a, /*neg_b=*/false, b,
      /*c_mod=*/(short)0, c, /*reuse_a=*/false, /*reuse_b=*/false);
}

// async 16B copy: global -> LDS, tracked by ASYNCcnt (no VGPR round trip)
__device__ __forceinline__ void async_copy_b128(void* lds_dst, const void* gsrc) {
  asm volatile("global_load_async_to_lds_b128 %0, %1, off"
               :: "v"((lds_void*)lds_dst), "v"(gsrc) : "memory");
}
__device__ __forceinline__ void wait_asynccnt0() {
  asm volatile("s_wait_asynccnt 0x0" ::: "memory");
}

// =====================================================================
// K1: tconv1 + GLU.   h1[b,o,t,n] = (conv_a + bias + x[b,o,t+2,n]) * sigmoid(conv_g + bias_g)
// =====================================================================
__global__ __launch_bounds__(256) void k_tconv1_glu(
    const float* __restrict__ x, const float* __restrict__ w1,
    const float* __restrict__ b1, float* __restrict__ h1)
{
  extern __shared__ float sm[];
  float* sX = sm;              // [192][64]   x tile, k = cc*3+i
  float* sW = sm + KCC * 64;   // [128][192]  tconv1 weights (o, cc, kt) flat

  const int tid = threadIdx.x;
  const int n0  = blockIdx.x * 64;
  const int t   = blockIdx.y;      // 0..9
  const int b   = blockIdx.z;

  for (int idx = tid; idx < 2 * CCH * KCC; idx += 256) sW[idx] = w1[idx];
  const size_t xb = (size_t)b * CCH * T0C * NN;
  for (int idx = tid; idx < KCC * 64; idx += 256) {
    int k = idx >> 6, n = idx & 63;
    int cc = k / 3, i = k - cc * 3;
    sX[idx] = x[xb + ((size_t)cc * T0C + t + i) * NN + n0 + n];
  }
  __syncthreads();

  const int n  = tid & 63;
  const int og = tid >> 6;       // 0..3 -> o = og*16 + j
  float accA[16], accG[16];
#pragma unroll
  for (int j = 0; j < 16; ++j) { accA[j] = 0.f; accG[j] = 0.f; }

  for (int k = 0; k < KCC; k += 4) {
    float x0 = sX[(k + 0) * 64 + n], x1 = sX[(k + 1) * 64 + n];
    float x2 = sX[(k + 2) * 64 + n], x3 = sX[(k + 3) * 64 + n];
#pragma unroll
    for (int j = 0; j < 16; ++j) {
      const float4 wa = *(const float4*)(sW + (og * 16 + j) * KCC + k);
      const float4 wg = *(const float4*)(sW + (CCH + og * 16 + j) * KCC + k);
      accA[j] += x0 * wa.x + x1 * wa.y + x2 * wa.z + x3 * wa.w;
      accG[j] += x0 * wg.x + x1 * wg.y + x2 * wg.z + x3 * wg.w;
    }
  }

#pragma unroll
  for (int j = 0; j < 16; ++j) {
    int o = og * 16 + j;
    float xin = x[xb + ((size_t)o * T0C + t + 2) * NN + n0 + n];
    float a = accA[j] + b1[o] + xin;
    float g = accG[j] + b1[CCH + o];
    float sg = 1.f / (1.f + __expf(-g));
    h1[((size_t)(b * CCH + o) * T1C + t) * NN + n0 + n] = a * sg;
  }
}

// =====================================================================
// K2: WMMA fp32 GEMM  P(640 x 6144) = A(640 x 2048) @ G(2048 x 6144)
// block tile 64(M) x 128(N), 8 waves, wave tile 32x32 = 2x2 WMMA tiles.
// Double-buffered LDS, staged with GLOBAL_LOAD_ASYNC_TO_LDS_B128.
// =====================================================================
#define A_STRIDE 20    // 16 used + pad (80B rows, 16B aligned, conflict-free frags)
#define B_STRIDE 136   // 128 used + pad (544B rows, 16B aligned, half-wave bank split)
#define A_TILE  (64 * A_STRIDE)   // 1280 floats
#define B_TILE  (16 * B_STRIDE)   // 2176 floats

__device__ __forceinline__ void stage_tile_async(
    const float* __restrict__ A, const float* __restrict__ G,
    float* As, float* Bs, int m0, int n0, int k0, int tid)
{
  // A tile 64x16: one b128 per thread
  const int lm = tid >> 2;            // 0..63
  const int lk = (tid & 3) << 2;      // 0,4,8,12
  async_copy_b128(As + lm * A_STRIDE + lk,
                  A + (size_t)(m0 + lm) * 2048 + k0 + lk);
  // B tile 16x128: two b128 per thread
#pragma unroll
  for (int r = 0; r < 2; ++r) {
    int linear = tid + (r << 8);      // 0..511 float4 slots
    int row = linear >> 5;            // 0..15
    int c4  = (linear & 31) << 2;     // 0..124
    async_copy_b128(Bs + row * B_STRIDE + c4,
                    G + (size_t)(k0 + row) * NKC + n0 + c4);
  }
}

__global__ __launch_bounds__(256) void k_gemm_scl(
    const float* __restrict__ A,   // h1 for this batch: 640 x 2048 row-major
    const float* __restrict__ G,   // g_kernel: 2048 x 6144 row-major
    float* __restrict__ P)         // 640 x 6144
{
  extern __shared__ float sm[];
  float* As0 = sm;
  float* As1 = sm + A_TILE;
  float* Bs0 = sm + 2 * A_TILE;
  float* Bs1 = sm + 2 * A_TILE + B_TILE;

  const int tid  = threadIdx.x;
  const int lane = tid & 31;
  const int wave = tid >> 5;
  const int wm   = wave >> 2;          // 0..1  (M)
  const int wn   = wave & 3;           // 0..3  (N)
  const int mrow = lane & 15;          // M row within 16x16 (also N col for B)
  const int koff = (lane >> 4) << 1;   // K pair select per ISA 16x4 layout
  const int n0   = blockIdx.x * 128;
  const int m0   = blockIdx.y * 64;

  v8f acc[2][2];
#pragma unroll
  for (int i = 0; i < 2; ++i)
#pragma unroll
    for (int j = 0; j < 2; ++j)
#pragma unroll
      for (int r = 0; r < 8; ++r) acc[i][j][r] = 0.f;

  // prologue: fetch tile 0
  stage_tile_async(A, G, As0, Bs0, m0, n0, 0, tid);

  for (int k0 = 0; k0 < 2048; k0 += 16) {
    float* Ac = (k0 & 16) ? As1 : As0;
    float* Bc = (k0 & 16) ? Bs1 : Bs0;

    wait_asynccnt0();     // my portion of the current tile has landed in LDS
    __syncthreads();      // everyone's portion landed; prev tile's reads retired

    if (k0 + 16 < 2048) { // overlap: fetch next tile into the other buffer
      float* An = (k0 & 16) ? As0 : As1;
      float* Bn = (k0 & 16) ? Bs0 : Bs1;
      stage_tile_async(A, G, An, Bn, m0, n0, k0 + 16, tid);
    }

#pragma unroll
    for (int kk = 0; kk < 16; kk += 4) {
      v2f a0 = *(const v2f*)(Ac + (wm * 32 + mrow) * A_STRIDE + kk + koff);
      v2f a1 = *(const v2f*)(Ac + (wm * 32 + 16 + mrow) * A_STRIDE + kk + koff);
      const int nb = wn * 32 + mrow;
      v2f b0, b1;
      b0.x = Bc[(kk + koff) * B_STRIDE + nb];
      b0.y = Bc[(kk + koff + 1) * B_STRIDE + nb];
      b1.x = Bc[(kk + koff) * B_STRIDE + nb + 16];
      b1.y = Bc[(kk + koff + 1) * B_STRIDE + nb + 16];
      acc[0][0] = wmma_f32_k4(a0, b0, acc[0][0]);
      acc[0][1] = wmma_f32_k4(a0, b1, acc[0][1]);
      acc[1][0] = wmma_f32_k4(a1, b0, acc[1][0]);
      acc[1][1] = wmma_f32_k4(a1, b1, acc[1][1]);
    }
  }

  // store D tiles: VGPR r -> M=r (lanes 0-15) / M=r+8 (lanes 16-31), N=lane&15
  const int halfsel = lane >> 4;
#pragma unroll
  for (int i = 0; i < 2; ++i)
#pragma unroll
    for (int j = 0; j < 2; ++j) {
      int mb = m0 + wm * 32 + i * 16 + halfsel * 8;
      int nn = n0 + wn * 32 + j * 16 + mrow;
#pragma unroll
      for (int r = 0; r < 8; ++r)
        P[(size_t)(mb + r) * NKC + nn] = acc[i][j][r];
    }
}

// =====================================================================
// K3: contract (c,k)->o, add bias + residual, ReLU.
// h2[o,t,m] = relu( sum_{c,k} P[(c*10+t)*6144 + k*2048 + m] * ws[c*192+k*64+o] + bs[o] + h1[o,t,m] )
// =====================================================================
__global__ __launch_bounds__(256) void k_scl_combine(
    const float* __restrict__ P, const float* __restrict__ h1b,
    const float* __restrict__ wsp, const float* __restrict__ bs,
    float* __restrict__ h2b)
{
  extern __shared__ float sm[];
  float* sP = sm;              // [192][64] : ck = c*3+k
  float* sW = sm + KCC * 64;   // [64*192]  : ws flat (c,k,o)

  const int tid = threadIdx.x;
  const int n0  = blockIdx.x * 64;
  const int t   = blockIdx.y;

  for (int idx = tid; idx < CCH * KCC; idx += 256) sW[idx] = wsp[idx];
  for (int idx = tid; idx < KCC * 64; idx += 256) {
    int ck = idx >> 6, m = idx & 63;
    int c = ck / 3, k = ck - c * 3;
    sP[idx] = P[((size_t)c * T1C + t) * NKC + (size_t)k * NN + n0 + m];
  }
  __syncthreads();

  const int m  = tid & 63;
  const int og = tid >> 6;
  float acc[16];
#pragma unroll
  for (int j = 0; j < 16; ++j) acc[j] = 0.f;

  for (int c = 0; c < CCH; ++c) {
#pragma unroll
    for (int k = 0; k < 3; ++k) {
      float pv = sP[(c * 3 + k) * 64 + m];
      const int base = c * KCC + k * 64 + og * 16;
#pragma unroll
      for (int q = 0; q < 4; ++q) {
        const float4 w4 = *(const float4*)(sW + base + q * 4);
        acc[q * 4 + 0] += pv * w4.x;
        acc[q * 4 + 1] += pv * w4.y;
        acc[q * 4 + 2] += pv * w4.z;
        acc[q * 4 + 3] += pv * w4.w;
      }
    }
  }

#pragma unroll
  for (int j = 0; j < 16; ++j) {
    int o = og * 16 + j;
    size_t off = ((size_t)o * T1C + t) * NN + n0 + m;
    float v = acc[j] + bs[o] + h1b[off];
    h2b[off] = fmaxf(v, 0.f);
  }
}

// =====================================================================
// K4: tconv2 + residual ReLU.  h3[b,o,t,n] = relu(conv(h2)+b2[o]+h2[b,o,t+2,n]), t in 0..7
// =====================================================================
__global__ __launch_bounds__(256) void k_tconv2_relu(
    const float* __restrict__ h2, const float* __restrict__ w2,
    const float* __restrict__ b2, float* __restrict__ h3)
{
  extern __shared__ float sm[];
  float* sX = sm;              // [192][64]
  float* sW = sm + KCC * 64;   // [64][192]

  const int tid = threadIdx.x;
  const int n0  = blockIdx.x * 64;
  const int t   = blockIdx.y;   // 0..7
  const int b   = blockIdx.z;

  for (int idx = tid; idx < CCH * KCC; idx += 256) sW[idx] = w2[idx];
  const size_t hb = (size_t)b * CCH * T1C * NN;
  for (int idx = tid; idx < KCC * 64; idx += 256) {
    int k = idx >> 6, n = idx & 63;
    int cc = k / 3, i = k - cc * 3;
    sX[idx] = h2[hb + ((size_t)cc * T1C + t + i) * NN + n0 + n];
  }
  __syncthreads();

  const int n  = tid & 63;
  const int og = tid >> 6;
  float acc[16];
#pragma unroll
  for (int j = 0; j < 16; ++j) acc[j] = 0.f;

  for (int k = 0; k < KCC; k += 4) {
    float x0 = sX[(k + 0) * 64 + n], x1 = sX[(k + 1) * 64 + n];
    float x2 = sX[(k + 2) * 64 + n], x3 = sX[(k + 3) * 64 + n];
#pragma unroll
    for (int j = 0; j < 16; ++j) {
      const float4 w4 = *(const float4*)(sW + (og * 16 + j) * KCC + k);
      acc[j] += x0 * w4.x + x1 * w4.y + x2 * w4.z + x3 * w4.w;
    }
  }

#pragma unroll
  for (int j = 0; j < 16; ++j) {
    int o = og * 16 + j;
    float xin = h2[hb + ((size_t)o * T1C + t + 2) * NN + n0 + n];
    float v = acc[j] + b2[o] + xin;
    h3[((size_t)(b * CCH + o) * T2C + t) * NN + n0 + n] = fmaxf(v, 0.f);
  }
}

// =====================================================================
// K5: LayerNorm over (N,C) per (b,t); output already in (B,C,T2,N) layout.
// =====================================================================
__global__ __launch_bounds__(256) void k_layernorm(
    const float* __restrict__ h3, const float* __restrict__ lnw,
    const float* __restrict__ lnb, float* __restrict__ out)
{
  __shared__ float red[256];
  __shared__ float red2[256];
  const int t = blockIdx.x;   // 0..7
  const int b = blockIdx.y;   // 0..15
  const int tid = threadIdx.x;
  const size_t baseB = (size_t)b * CCH * T2C * NN;
  const int total = CCH * NN;   // 131072

  float s = 0.f, s2 = 0.f;
  for (int idx = tid; idx < total; idx += 256) {
    int c = idx >> 11, n = idx & (NN - 1);
    float v = h3[baseB + ((size_t)c * T2C + t) * NN + n];
    s += v; s2 += v * v;
  }
  red[tid] = s; red2[tid] = s2;
  __syncthreads();
  for (int off = 128; off > 0; off >>= 1) {
    if (tid < off) { red[tid] += red[tid + off]; red2[tid] += red2[tid + off]; }
    __syncthreads();
  }
  const float inv = 1.f / (float)total;
  float mu  = red[0] * inv;
  float var = red2[0] * inv - mu * mu;
  float rsig = rsqrtf(var + 1e-5f);

  for (int idx = tid; idx < total; idx += 256) {
    int c = idx >> 11, n = idx & (NN - 1);
    size_t off = baseB + ((size_t)c * T2C + t) * NN + n;
    float v = h3[off];
    out[off] = (v - mu) * rsig * lnw[(size_t)n * CCH + c] + lnb[(size_t)n * CCH + c];
  }
}

// =====================================================================
extern "C" void kernel_launch(void* const* d_in, const int* in_sizes, int n_in,
                              void* d_out, int out_size, void* d_ws, size_t ws_size,
                              hipStream_t stream) {
  const float* x   = (const float*)d_in[0];
  const float* gk  = (const float*)d_in[1];
  const float* wsp = (const float*)d_in[2];
  const float* bs  = (const float*)d_in[3];
  const float* w1  = (const float*)d_in[4];
  const float* b1  = (const float*)d_in[5];
  const float* w2  = (const float*)d_in[6];
  const float* b2  = (const float*)d_in[7];
  const float* lnw = (const float*)d_in[8];
  const float* lnb = (const float*)d_in[9];
  float* out = (float*)d_out;

  // workspace layout (floats): h1 | h2 | h3 | P   (~251 MB total)
  float* h1 = (float*)d_ws;                                 // 16*64*10*2048
  float* h2 = h1 + (size_t)BB * CCH * T1C * NN;             // 16*64*10*2048
  float* h3 = h2 + (size_t)BB * CCH * T1C * NN;             // 16*64*8*2048
  float* P  = h3 + (size_t)BB * CCH * T2C * NN;             // 640*6144 (per-batch, reused)

  // K1: tconv1 + GLU
  k_tconv1_glu<<<dim3(NN / 64, T1C, BB), 256,
                 (KCC * 64 + 2 * CCH * KCC) * sizeof(float), stream>>>(x, w1, b1, h1);

  // K2/K3 per batch: big WMMA GEMM (async-LDS double buffered) then (c,k) contraction
  for (int b = 0; b < BB; ++b) {
    const float* h1b = h1 + (size_t)b * CCH * T1C * NN;
    float* h2b       = h2 + (size_t)b * CCH * T1C * NN;
    k_gemm_scl<<<dim3(NKC / 128, 640 / 64), 256,
                 2 * (A_TILE + B_TILE) * sizeof(float), stream>>>(h1b, gk, P);
    k_scl_combine<<<dim3(NN / 64, T1C), 256,
                    (KCC * 64 + CCH * KCC) * sizeof(float), stream>>>(P, h1b, wsp, bs, h2b);
  }

  // K4: tconv2 + residual ReLU
  k_tconv2_relu<<<dim3(NN / 64, T2C, BB), 256,
                  (KCC * 64 + CCH * KCC) * sizeof(float), stream>>>(h2, w2, b2, h3);

  // K5: LayerNorm
  k_layernorm<<<dim3(T2C, BB), 256, 0, stream>>>(h3, lnw, lnb, out);
}